// GATModel_30605936951469
// MI455X (gfx1250) — compile-verified
//
#include <hip/hip_runtime.h>
#include <math.h>

// GATv2 (2 layers) for MI455X / gfx1250.
// N=100000 nodes, E=1.6M edges, F_IN=128, F_E=8, H=4, C=16 (H*C=64), T=1.
//
// Dense transforms use V_WMMA_F32_16X16X4_F32 (fp32 matrix cores — the model
// is fp32, so no precision downcast). Edge phase is gather/scatter bound and
// lives in the 192MB L2 (xl1+xr1+acc ~ 80MB resident), implemented with float
// atomics + CAS-based atomic max for the segment softmax.

typedef __attribute__((ext_vector_type(2))) float v2f;
typedef __attribute__((ext_vector_type(8))) float v8f;

#define F_IN 128
#define HC   64      // H*C
#define NH   4       // heads layer 1
#define CC   16      // channels per head
#define FE   8       // edge feature dim
#define NEG_SLOPE 0.2f

static __device__ __forceinline__ float leakyrelu(float v) {
    return v > 0.f ? v : NEG_SLOPE * v;
}

static __device__ __forceinline__ void atomicMaxF(float* addr, float val) {
    int* ai = (int*)addr;
    int cur = __float_as_int(*addr);
    while (__int_as_float(cur) < val) {
        int prev = atomicCAS(ai, cur, __float_as_int(val));
        if (prev == cur) break;
        cur = prev;
    }
}

// ---------------------------------------------------------------------------
// generic fill
__global__ void k_fill_f32(float* __restrict__ p, long long n, float v) {
    long long i = (long long)blockIdx.x * blockDim.x + threadIdx.x;
    if (i < n) p[i] = v;
}

// ---------------------------------------------------------------------------
// Layer-1 node transform: XL = X@Wl + bl, XR = X@Wr + br  (N x 128 @ 128 x 64)
// One wave per 16-row tile. 8 output column tiles (4 for Wl, 4 for Wr),
// each a v8f accumulator. K loop in steps of 4 using fp32 WMMA.
//
// Fragment layouts (CDNA5 ISA 7.12.2, wave32):
//   A 16x4 f32 : lane L(0..15) row=L, lanes 16..31 same rows; VGPR j holds
//                K = k0 + 2*(lane>=16) + j
//   B 4x16 f32 : lane gives col (lane&15); VGPR j holds row K = k0 + 2*half + j
//   C 16x16 f32: VGPR v -> row v + 8*(lane>=16), col = lane&15
__global__ __launch_bounds__(32)
void k_node_gemm_l1(const float* __restrict__ X,
                    const float* __restrict__ Wl, const float* __restrict__ bl,
                    const float* __restrict__ Wr, const float* __restrict__ br,
                    float* __restrict__ XL, float* __restrict__ XR, int N_)
{
    const int m0   = blockIdx.x * 16;
    const int lane = threadIdx.x & 31;
    const int half = lane >> 4;   // 0 or 1
    const int l15  = lane & 15;

    v8f acc[8] = {};

    const int arow = (m0 + l15 < N_) ? (m0 + l15) : (N_ - 1);
    const float* __restrict__ xrow = X + (size_t)arow * F_IN;

    for (int k0 = 0; k0 < F_IN; k0 += 4) {
        v2f a;
        a.x = xrow[k0 + 2 * half + 0];
        a.y = xrow[k0 + 2 * half + 1];
#pragma unroll
        for (int t = 0; t < 8; ++t) {
            const float* __restrict__ W = (t < 4) ? Wl : Wr;
            const int n0 = (t & 3) * 16;
            v2f b;
            b.x = W[(size_t)(k0 + 2 * half + 0) * HC + n0 + l15];
            b.y = W[(size_t)(k0 + 2 * half + 1) * HC + n0 + l15];
            acc[t] = __builtin_amdgcn_wmma_f32_16x16x4_f32(
                false, a, false, b, (short)0, acc[t], false, false);
        }
    }

#pragma unroll
    for (int t = 0; t < 8; ++t) {
        float* __restrict__ O   = (t < 4) ? XL : XR;
        const float* __restrict__ bias = (t < 4) ? bl : br;
        const int n0 = (t & 3) * 16;
        const float bv = bias[n0 + l15];
#pragma unroll
        for (int v = 0; v < 8; ++v) {
            const int row = m0 + v + 8 * half;
            if (row < N_) O[(size_t)row * HC + n0 + l15] = acc[t][v] + bv;
        }
    }
}

// ---------------------------------------------------------------------------
// sum of edge_attr over edges -> emean[8] (raw sums; scaled by 1/E at use)
__global__ void k_edge_attr_mean(const float* __restrict__ ea,
                                 float* __restrict__ emean, int E_)
{
    float s[FE];
#pragma unroll
    for (int f = 0; f < FE; ++f) s[f] = 0.f;
    for (int i = blockIdx.x * blockDim.x + threadIdx.x; i < E_;
         i += gridDim.x * blockDim.x) {
#pragma unroll
        for (int f = 0; f < FE; ++f) s[f] += ea[(size_t)i * FE + f];
    }
#pragma unroll
    for (int f = 0; f < FE; ++f)
        for (int o = 16; o > 0; o >>= 1) s[f] += __shfl_xor(s[f], o, 32);
    if ((threadIdx.x & 31) == 0) {
#pragma unroll
        for (int f = 0; f < FE; ++f) atomicAdd(&emean[f], s[f]);
    }
}

// ---------------------------------------------------------------------------
// Layer-1 attention pass A: logits[e,h] + running segment max (atomic CAS-max)
__global__ void k_edge_logits_l1(const int* __restrict__ src,
                                 const int* __restrict__ dst,
                                 const float* __restrict__ eattr,
                                 const float* __restrict__ emean, float inv_e,
                                 const float* __restrict__ We1,
                                 const float* __restrict__ att1,
                                 const float* __restrict__ XL,
                                 const float* __restrict__ XR,
                                 float* __restrict__ logits,
                                 float* __restrict__ m1, int E_, int N_)
{
    const int idx = blockIdx.x * blockDim.x + threadIdx.x;
    const int total = (E_ + N_) * NH;
    if (idx >= total) return;
    const int h = idx & (NH - 1);
    const int i = idx >> 2;

    int s, d;
    float ef[FE];
    if (i < E_) {
        s = src[i]; d = dst[i];
#pragma unroll
        for (int f = 0; f < FE; ++f) ef[f] = eattr[(size_t)i * FE + f];
    } else {
        s = d = i - E_;
#pragma unroll
        for (int f = 0; f < FE; ++f) ef[f] = emean[f] * inv_e;
    }

    const float* __restrict__ xl = XL + (size_t)s * HC + h * CC;
    const float* __restrict__ xr = XR + (size_t)d * HC + h * CC;
    const float* __restrict__ we = We1 + h * CC;   // We1[f*64 + h*16 + c]
    const float* __restrict__ at = att1 + h * CC;

    float logit = 0.f;
#pragma unroll
    for (int c = 0; c < CC; ++c) {
        float ee = 0.f;
#pragma unroll
        for (int f = 0; f < FE; ++f) ee += ef[f] * we[f * HC + c];
        logit += at[c] * leakyrelu(xl[c] + xr[c] + ee);
    }
    logits[idx] = logit;
    atomicMaxF(&m1[d * NH + h], logit);
}

// pass B: p = exp(logit - max); denom += p
__global__ void k_edge_exp_l1(const int* __restrict__ dst,
                              float* __restrict__ logits,
                              const float* __restrict__ m1,
                              float* __restrict__ d1, int E_, int N_)
{
    const int idx = blockIdx.x * blockDim.x + threadIdx.x;
    const int total = (E_ + N_) * NH;
    if (idx >= total) return;
    const int h = idx & (NH - 1);
    const int i = idx >> 2;
    const int d = (i < E_) ? dst[i] : (i - E_);
    const float p = __expf(logits[idx] - m1[d * NH + h]);
    logits[idx] = p;
    atomicAdd(&d1[d * NH + h], p);
}

// pass C: scatter-add alpha * xl[src] into destination accumulator
__global__ void k_edge_agg_l1(const int* __restrict__ src,
                              const int* __restrict__ dst,
                              const float* __restrict__ p,
                              const float* __restrict__ d1,
                              const float* __restrict__ XL,
                              float* __restrict__ hacc, int E_, int N_)
{
    const int idx = blockIdx.x * blockDim.x + threadIdx.x;
    const int total = (E_ + N_) * NH;
    if (idx >= total) return;
    const int h = idx & (NH - 1);
    const int i = idx >> 2;
    int s, d;
    if (i < E_) { s = src[i]; d = dst[i]; } else { s = d = i - E_; }
    const float alpha = p[idx] / d1[d * NH + h];
    const float* __restrict__ xl = XL + (size_t)s * HC + h * CC;
    float* __restrict__ out = hacc + (size_t)d * HC + h * CC;
#pragma unroll
    for (int c = 0; c < CC; ++c) atomicAdd(&out[c], alpha * xl[c]);
}

// h = elu(acc + b1)  (in place)
__global__ void k_finalize_l1(float* __restrict__ hacc,
                              const float* __restrict__ b1, long long n64)
{
    const long long i = (long long)blockIdx.x * blockDim.x + threadIdx.x;
    if (i >= n64) return;
    const float v = hacc[i] + b1[i & (HC - 1)];
    hacc[i] = v > 0.f ? v : (__expf(v) - 1.f);
}

// ---------------------------------------------------------------------------
// Layer-2 node transform (T=1 -> GEMV): xl2 = h@Wl2 + bl2, xr2 = h@Wr2 + br2
__global__ void k_node_gemv_l2(const float* __restrict__ Hb,
                               const float* __restrict__ Wl2, const float* __restrict__ bl2,
                               const float* __restrict__ Wr2, const float* __restrict__ br2,
                               float* __restrict__ XL2, float* __restrict__ XR2, int N_)
{
    const int n = blockIdx.x * blockDim.x + threadIdx.x;
    if (n >= N_) return;
    const float* __restrict__ hrow = Hb + (size_t)n * HC;
    float sl = 0.f, sr = 0.f;
#pragma unroll
    for (int j = 0; j < HC; ++j) {
        const float hv = hrow[j];
        sl += hv * Wl2[j];
        sr += hv * Wr2[j];
    }
    XL2[n] = sl + bl2[0];
    XR2[n] = sr + br2[0];
}

__global__ void k_edge_logits_l2(const int* __restrict__ src,
                                 const int* __restrict__ dst,
                                 const float* __restrict__ eattr,
                                 const float* __restrict__ emean, float inv_e,
                                 const float* __restrict__ We2,
                                 const float* __restrict__ att2,
                                 const float* __restrict__ XL2,
                                 const float* __restrict__ XR2,
                                 float* __restrict__ logits2,
                                 float* __restrict__ m2, int E_, int N_)
{
    const int i = blockIdx.x * blockDim.x + threadIdx.x;
    if (i >= E_ + N_) return;
    int s, d;
    float ee = 0.f;
    if (i < E_) {
        s = src[i]; d = dst[i];
#pragma unroll
        for (int f = 0; f < FE; ++f) ee += eattr[(size_t)i * FE + f] * We2[f];
    } else {
        s = d = i - E_;
#pragma unroll
        for (int f = 0; f < FE; ++f) ee += emean[f] * inv_e * We2[f];
    }
    const float logit = att2[0] * leakyrelu(XL2[s] + XR2[d] + ee);
    logits2[i] = logit;
    atomicMaxF(&m2[d], logit);
}

__global__ void k_edge_exp_l2(const int* __restrict__ dst,
                              float* __restrict__ logits2,
                              const float* __restrict__ m2,
                              float* __restrict__ d2, int E_, int N_)
{
    const int i = blockIdx.x * blockDim.x + threadIdx.x;
    if (i >= E_ + N_) return;
    const int d = (i < E_) ? dst[i] : (i - E_);
    const float p = __expf(logits2[i] - m2[d]);
    logits2[i] = p;
    atomicAdd(&d2[d], p);
}

__global__ void k_edge_agg_l2(const int* __restrict__ src,
                              const int* __restrict__ dst,
                              const float* __restrict__ p,
                              const float* __restrict__ d2,
                              const float* __restrict__ XL2,
                              float* __restrict__ acc2, int E_, int N_)
{
    const int i = blockIdx.x * blockDim.x + threadIdx.x;
    if (i >= E_ + N_) return;
    int s, d;
    if (i < E_) { s = src[i]; d = dst[i]; } else { s = d = i - E_; }
    atomicAdd(&acc2[d], (p[i] / d2[d]) * XL2[s]);
}

__global__ void k_finalize_l2(const float* __restrict__ acc2,
                              const float* __restrict__ b2,
                              float* __restrict__ out, int N_)
{
    const int n = blockIdx.x * blockDim.x + threadIdx.x;
    if (n < N_) out[n] = acc2[n] + b2[0];
}

// ---------------------------------------------------------------------------
static inline int cdiv(long long a, long long b) { return (int)((a + b - 1) / b); }

extern "C" void kernel_launch(void* const* d_in, const int* in_sizes, int n_in,
                              void* d_out, int out_size, void* d_ws, size_t ws_size,
                              hipStream_t stream)
{
    const float* x     = (const float*)d_in[0];
    const int*   eidx  = (const int*)  d_in[1];
    const float* eattr = (const float*)d_in[2];
    const float* Wl1   = (const float*)d_in[3];
    const float* bl1   = (const float*)d_in[4];
    const float* Wr1   = (const float*)d_in[5];
    const float* br1   = (const float*)d_in[6];
    const float* We1   = (const float*)d_in[7];
    const float* att1  = (const float*)d_in[8];
    const float* b1    = (const float*)d_in[9];
    const float* Wl2   = (const float*)d_in[10];
    const float* bl2   = (const float*)d_in[11];
    const float* Wr2   = (const float*)d_in[12];
    const float* br2   = (const float*)d_in[13];
    const float* We2   = (const float*)d_in[14];
    const float* att2  = (const float*)d_in[15];
    const float* b2    = (const float*)d_in[16];

    const int N  = in_sizes[0] / F_IN;
    const int E  = in_sizes[1] / 2;
    const int Ea = E + N;
    const int* src = eidx;        // edge_index[0, :]
    const int* dst = eidx + E;    // edge_index[1, :]
    const float inv_e = 1.0f / (float)E;

    // workspace layout (floats)
    float* ws = (float*)d_ws;
    size_t off = 0;
    float* xl1   = ws + off; off += (size_t)N * HC;
    float* xr1   = ws + off; off += (size_t)N * HC;
    float* hbuf  = ws + off; off += (size_t)N * HC;   // attention acc, then elu'd h
    float* lg1   = ws + off; off += (size_t)Ea * NH;  // logits -> p
    float* m1    = ws + off; off += (size_t)N * NH;
    float* d1    = ws + off; off += (size_t)N * NH;
    float* xl2   = ws + off; off += (size_t)N;
    float* xr2   = ws + off; off += (size_t)N;
    float* lg2   = ws + off; off += (size_t)Ea;
    float* m2    = ws + off; off += (size_t)N;
    float* d2    = ws + off; off += (size_t)N;
    float* acc2  = ws + off; off += (size_t)N;
    float* emean = ws + off; off += 16;
    (void)ws_size;

    const int TB = 256;
    const float NEG_INF = -__builtin_huge_valf();

    // init scratch
    k_fill_f32<<<cdiv((long long)N * HC, TB), TB, 0, stream>>>(hbuf, (long long)N * HC, 0.f);
    k_fill_f32<<<cdiv((long long)N * NH, TB), TB, 0, stream>>>(m1, (long long)N * NH, NEG_INF);
    k_fill_f32<<<cdiv((long long)N * NH, TB), TB, 0, stream>>>(d1, (long long)N * NH, 0.f);
    k_fill_f32<<<cdiv(N, TB), TB, 0, stream>>>(m2, N, NEG_INF);
    k_fill_f32<<<cdiv(N, TB), TB, 0, stream>>>(d2, N, 0.f);
    k_fill_f32<<<cdiv(N, TB), TB, 0, stream>>>(acc2, N, 0.f);
    k_fill_f32<<<1, 32, 0, stream>>>(emean, 16, 0.f);

    // dense transforms (WMMA f32) + self-loop edge-attr mean
    k_node_gemm_l1<<<cdiv(N, 16), 32, 0, stream>>>(x, Wl1, bl1, Wr1, br1, xl1, xr1, N);
    k_edge_attr_mean<<<512, TB, 0, stream>>>(eattr, emean, E);

    // layer-1 attention (segment softmax, 3 passes over (edge, head))
    const int tot1 = Ea * NH;
    k_edge_logits_l1<<<cdiv(tot1, TB), TB, 0, stream>>>(src, dst, eattr, emean, inv_e,
                                                        We1, att1, xl1, xr1, lg1, m1, E, N);
    k_edge_exp_l1<<<cdiv(tot1, TB), TB, 0, stream>>>(dst, lg1, m1, d1, E, N);
    k_edge_agg_l1<<<cdiv(tot1, TB), TB, 0, stream>>>(src, dst, lg1, d1, xl1, hbuf, E, N);
    k_finalize_l1<<<cdiv((long long)N * HC, TB), TB, 0, stream>>>(hbuf, b1, (long long)N * HC);

    // layer 2
    k_node_gemv_l2<<<cdiv(N, TB), TB, 0, stream>>>(hbuf, Wl2, bl2, Wr2, br2, xl2, xr2, N);
    k_edge_logits_l2<<<cdiv(Ea, TB), TB, 0, stream>>>(src, dst, eattr, emean, inv_e,
                                                      We2, att2, xl2, xr2, lg2, m2, E, N);
    k_edge_exp_l2<<<cdiv(Ea, TB), TB, 0, stream>>>(dst, lg2, m2, d2, E, N);
    k_edge_agg_l2<<<cdiv(Ea, TB), TB, 0, stream>>>(src, dst, lg2, d2, xl2, acc2, E, N);
    k_finalize_l2<<<cdiv(N, TB), TB, 0, stream>>>(acc2, b2, (float*)d_out, N);
}